// HeteroSAGE_24575802868492
// MI455X (gfx1250) — compile-verified
//
#include <hip/hip_runtime.h>

// HeteroSAGE forward on gfx1250 (MI455X, wave32).
// Per layer: scatter-mean (1/deg folded into contributions; float atomics
// resolve in the 192MB L2 since node arrays are <=102MB) -> fused dual-GEMM
// relu(agg@Wl + x@Wr + b) on the matrix pipe via V_WMMA_F32_16X16X4_F32.
// Weights are repacked once per layer into k-pair-interleaved float2 form so
// every B fragment is a single global_load_b64 at an immediate offset.

typedef __attribute__((ext_vector_type(2))) float v2f;
typedef __attribute__((ext_vector_type(8))) float v8f;

// ---------------------------------------------------------------- counts / inv
__global__ void hs_count_kernel(const int* __restrict__ srcIdx,
                                const int* __restrict__ dstIdx,
                                float* __restrict__ cntSrc,
                                float* __restrict__ cntDst, int E) {
    int e = blockIdx.x * blockDim.x + threadIdx.x;
    if (e < E) {
        atomicAdd(&cntSrc[srcIdx[e]], 1.0f);
        atomicAdd(&cntDst[dstIdx[e]], 1.0f);
    }
}

__global__ void hs_inv_kernel(float* __restrict__ c, int n) {
    int i = blockIdx.x * blockDim.x + threadIdx.x;
    if (i < n) c[i] = 1.0f / fmaxf(c[i], 1.0f);
}

// ------------------------------------------------------------- scatter (mean)
// One thread = one float4 chunk of one edge's gathered feature row.
// agg[target] += x[gather] * inv_count[target]   (mean folded in)
__global__ void hs_scatter_kernel(const float* __restrict__ x,
                                  const int* __restrict__ gatherIdx,
                                  const int* __restrict__ targetIdx,
                                  const float* __restrict__ inv,
                                  float* __restrict__ agg,
                                  int E, int log2d) {
    const int chunksPerEdge = 1 << (log2d - 2);
    const int total = E << (log2d - 2);
    int idx = blockIdx.x * blockDim.x + threadIdx.x;
    if (idx >= total) return;
    int e = idx >> (log2d - 2);
    int c = idx & (chunksPerEdge - 1);
    int g = gatherIdx[e];
    int t = targetIdx[e];
    float s = inv[t];
    const float4 v = *((const float4*)(x + ((long long)g << log2d)) + c);
    float* ar = agg + ((long long)t << log2d) + (c << 2);
    atomicAdd(ar + 0, v.x * s);
    atomicAdd(ar + 1, v.y * s);
    atomicAdd(ar + 2, v.z * s);
    atomicAdd(ar + 3, v.w * s);
}

// ------------------------------------------------- weight repack (k-pair AoS)
// Wp (as float2): Wp[p*H + n] = { W[2p][n], W[2p+1][n] }  for p in [0, K/2)
// -> the B fragment for WMMA lane (needs rows koff, koff+1 at col n) becomes
//    one aligned 8-byte load.
__global__ void hs_pack_kernel(const float* __restrict__ W,
                               float2* __restrict__ Wp, int K, int H) {
    int i = blockIdx.x * blockDim.x + threadIdx.x;  // over (K/2)*H
    if (i >= (K >> 1) * H) return;
    int p = i / H;
    int n = i - p * H;
    float2 v;
    v.x = W[(2 * p) * H + n];
    v.y = W[(2 * p + 1) * H + n];
    Wp[i] = v;
}

// ----------------------------------------------------- fused dual-GEMM + ReLU
// out[M,H] = relu(agg[M,K] @ Wl[K,H] + x[M,K] @ Wr[K,H] + bias[H])
// One wave -> 16x64 output tile, 4 accumulators of V_WMMA_F32_16X16X4_F32.
// A frag (16x4 f32): lanes 0-15 = rows; lane half selects K pair {0,1}/{2,3}
//   -> one aligned float2 load per lane per input matrix.
// B frag (4x16 f32): one float2 (packed) load per lane per sub-tile.
// C/D frag: VGPR r -> M = r + 8*(lane>=16), N = lane&15.
__global__ void __launch_bounds__(256)
hs_gemm_kernel(const float* __restrict__ agg, const float* __restrict__ x,
               const v2f* __restrict__ WlP, const v2f* __restrict__ WrP,
               const float* __restrict__ bias, float* __restrict__ out,
               int nRows, int K, int H) {
    const int lane = threadIdx.x & 31;
    const int wave = threadIdx.x >> 5;
    const int mTiles = nRows >> 4;     // 6250 for N=100000 (exact)
    const int nGroups = H >> 6;        // 4 groups of 64 cols for H=256
    const int wt = blockIdx.x * 8 + wave;
    if (wt >= mTiles * nGroups) return;           // wave-uniform exit

    const int tileM = wt / nGroups;
    const int ng = wt - tileM * nGroups;
    const int n0 = ng << 6;
    const int halfSel = lane >> 4;     // 0 -> K pair {0,1}; 1 -> {2,3}
    const int mLane = lane & 15;
    const int rowA = (tileM << 4) + mLane;

    // A pointers advance by 2 v2f (4 floats) per k-step.
    const v2f* aggP = (const v2f*)(agg + (long long)rowA * K + 2 * halfSel);
    const v2f* xP   = (const v2f*)(x   + (long long)rowA * K + 2 * halfSel);
    // Packed-B pointers: element index p*H + col; p advances by 2 per k-step.
    const v2f* wlp = WlP + (long long)halfSel * H + n0 + mLane;
    const v2f* wrp = WrP + (long long)halfSel * H + n0 + mLane;
    const int bStep = 2 * H;           // v2f elements per k-step

    v8f acc0 = {0,0,0,0,0,0,0,0};
    v8f acc1 = {0,0,0,0,0,0,0,0};
    v8f acc2 = {0,0,0,0,0,0,0,0};
    v8f acc3 = {0,0,0,0,0,0,0,0};

    for (int k = 0; k < K; k += 4) {
        v2f aA = aggP[0];
        v2f aX = xP[0];
        v2f bl0 = wlp[0], bl1 = wlp[16], bl2 = wlp[32], bl3 = wlp[48];
        v2f br0 = wrp[0], br1 = wrp[16], br2 = wrp[32], br3 = wrp[48];

        acc0 = __builtin_amdgcn_wmma_f32_16x16x4_f32(false, aA, false, bl0, (short)0, acc0, false, false);
        acc1 = __builtin_amdgcn_wmma_f32_16x16x4_f32(false, aA, false, bl1, (short)0, acc1, false, false);
        acc2 = __builtin_amdgcn_wmma_f32_16x16x4_f32(false, aA, false, bl2, (short)0, acc2, false, false);
        acc3 = __builtin_amdgcn_wmma_f32_16x16x4_f32(false, aA, false, bl3, (short)0, acc3, false, false);
        acc0 = __builtin_amdgcn_wmma_f32_16x16x4_f32(false, aX, false, br0, (short)0, acc0, false, false);
        acc1 = __builtin_amdgcn_wmma_f32_16x16x4_f32(false, aX, false, br1, (short)0, acc1, false, false);
        acc2 = __builtin_amdgcn_wmma_f32_16x16x4_f32(false, aX, false, br2, (short)0, acc2, false, false);
        acc3 = __builtin_amdgcn_wmma_f32_16x16x4_f32(false, aX, false, br3, (short)0, acc3, false, false);

        aggP += 2;
        xP   += 2;
        wlp  += bStep;
        wrp  += bStep;
    }

    // Epilogue: bias + ReLU + store per documented C/D layout.
    const int colBase = n0 + mLane;
    const int rowBase = (tileM << 4) + (halfSel << 3);
    const float bv0 = bias[colBase];
    const float bv1 = bias[colBase + 16];
    const float bv2 = bias[colBase + 32];
    const float bv3 = bias[colBase + 48];
#pragma unroll
    for (int r = 0; r < 8; ++r) {
        float* orow = out + (long long)(rowBase + r) * H + colBase;
        orow[0]  = fmaxf(acc0[r] + bv0, 0.0f);
        orow[16] = fmaxf(acc1[r] + bv1, 0.0f);
        orow[32] = fmaxf(acc2[r] + bv2, 0.0f);
        orow[48] = fmaxf(acc3[r] + bv3, 0.0f);
    }
}

// ---------------------------------------------------------------- orchestration
extern "C" void kernel_launch(void* const* d_in, const int* in_sizes, int n_in,
                              void* d_out, int out_size, void* d_ws, size_t ws_size,
                              hipStream_t stream) {
    const float* x_human = (const float*)d_in[0];
    const float* x_bact  = (const float*)d_in[1];
    const int*   edge    = (const int*)d_in[2];
    const float* h1_Wl = (const float*)d_in[3];
    const float* h1_Wr = (const float*)d_in[4];
    const float* h1_b  = (const float*)d_in[5];
    const float* h2_Wl = (const float*)d_in[6];
    const float* h2_Wr = (const float*)d_in[7];
    const float* h2_b  = (const float*)d_in[8];
    const float* b1_Wl = (const float*)d_in[9];
    const float* b1_Wr = (const float*)d_in[10];
    const float* b1_b  = (const float*)d_in[11];
    const float* b2_Wl = (const float*)d_in[12];
    const float* b2_Wr = (const float*)d_in[13];
    const float* b2_b  = (const float*)d_in[14];

    const int DIN = 128;
    const int H   = in_sizes[5];          // 256
    const int N   = in_sizes[0] / DIN;    // 100000
    const int E   = in_sizes[2] / 2;      // 1600000
    const int* srcIdx = edge;             // edge_index[0]
    const int* dstIdx = edge + E;         // edge_index[1]

    // Workspace (floats):
    //   inv_src[N] | inv_dst[N] | agg[N*256] | tmp[N*256] | WlP[K*H] | WrP[K*H]
    float* inv_src = (float*)d_ws;
    float* inv_dst = inv_src + N;
    float* agg     = inv_dst + N;
    float* tmp     = agg + (size_t)N * 256;
    float* wlPack  = tmp + (size_t)N * 256;
    float* wrPack  = wlPack + (size_t)256 * H;

    float* out_h = (float*)d_out;
    float* out_b = out_h + (size_t)N * H;

    // Degrees -> inverse mean scale (both branches at once).
    hipMemsetAsync(inv_src, 0, (size_t)2 * N * sizeof(float), stream);
    hs_count_kernel<<<(E + 255) / 256, 256, 0, stream>>>(srcIdx, dstIdx, inv_src, inv_dst, E);
    hs_inv_kernel<<<(2 * N + 255) / 256, 256, 0, stream>>>(inv_src, 2 * N);

    auto run_layer = [&](const float* xin, const int* gIdx, const int* tIdx,
                         const float* inv, const float* Wl, const float* Wr,
                         const float* bvec, float* o, int d) {
        const int log2d = (d == 128) ? 7 : 8;
        hipMemsetAsync(agg, 0, (size_t)N * d * sizeof(float), stream);
        const int work = E << (log2d - 2);
        hs_scatter_kernel<<<(work + 255) / 256, 256, 0, stream>>>(
            xin, gIdx, tIdx, inv, agg, E, log2d);
        const int packN = (d >> 1) * H;
        hs_pack_kernel<<<(packN + 255) / 256, 256, 0, stream>>>(Wl, (float2*)wlPack, d, H);
        hs_pack_kernel<<<(packN + 255) / 256, 256, 0, stream>>>(Wr, (float2*)wrPack, d, H);
        const int totalWaves = (N / 16) * (H / 64);
        hs_gemm_kernel<<<(totalWaves + 7) / 8, 256, 0, stream>>>(
            agg, xin, (const v2f*)wlPack, (const v2f*)wrPack, bvec, o, N, d, H);
    };

    // Human branch: rev edges -> gather at dst, scatter to src, degree over src.
    run_layer(x_human, dstIdx, srcIdx, inv_src, h1_Wl, h1_Wr, h1_b, tmp,   DIN);
    run_layer(tmp,     dstIdx, srcIdx, inv_src, h2_Wl, h2_Wr, h2_b, out_h, H);
    // Bacterial branch: fwd edges -> gather at src, scatter to dst, degree over dst.
    run_layer(x_bact,  srcIdx, dstIdx, inv_dst, b1_Wl, b1_Wr, b1_b, tmp,   DIN);
    run_layer(tmp,     srcIdx, dstIdx, inv_dst, b2_Wl, b2_Wr, b2_b, out_b, H);
}